// PerfusionExcitation_35897336659978
// MI455X (gfx1250) — compile-verified
//
#include <hip/hip_runtime.h>
#include <hip/hip_bf16.h>
#include <math.h>

// ---------------------------------------------------------------------------
// feats_ts: [B=4, Ttot=8, C=64, H=64, W=64] fp32; P=21, dilation 2, PAD=20
// out: [4, 8, 64, 64, 64] fp32 (last time frame zero)
// ---------------------------------------------------------------------------
#define PATCH   21
#define PP      441          // P*P channels
#define CC      64
#define HH      64
#define WW      64
#define TT      7            // time pairs
#define BB      4
#define NPAIR   28           // BB*TT
#define HW      4096         // HH*WW
#define BT_STR  68           // LDS stride for Bt[s][c] (even -> b64 aligned, 4-bank lane stride)
#define G_STR   105          // LDS stride for G rows
#define KP      448          // K=441 padded to 112 k-steps of 4
#define CBT_STR 452          // conv LDS stride Bt[pix][p] (even, 4-bank lane stride)

typedef __attribute__((ext_vector_type(2))) float v2f;
typedef __attribute__((ext_vector_type(8))) float v8f;

// ---------------------------------------------------------------------------
// Kernel 1: correlation via per-(n, dy, h) GEMM  G[w,s] = sum_c f1[c,w]*f2pad[c,s]
// Grid: (h=64, p=21, n=28), block = 128 (4 waves; wave m owns w-tile [16m,16m+16))
// f2 row staged transposed+padded in LDS so the WMMA loop is branch-free.
// ---------------------------------------------------------------------------
__global__ __launch_bounds__(128)
void corr_wmma_kernel(const float* __restrict__ feats, float* __restrict__ cor)
{
    const int h = blockIdx.x;
    const int p = blockIdx.y;
    const int n = blockIdx.z;
    const int b = n / TT, t = n % TT;
    const int r = h + 2 * p - 20;              // source row in f2
    const int tid  = threadIdx.x;
    const int lane = tid & 31;
    const int wave = tid >> 5;                 // M-tile index 0..3

    // cor layout: [n][pq][h][w], pq = p*21+q
    float* corRow = cor + (((size_t)n * PP + (size_t)p * PATCH) * HH + h) * WW;

    if (r < 0 || r >= HH) {                    // whole dy-row out of range -> zeros
        for (int e = tid; e < PATCH * WW; e += 128) {
            int q = e >> 6, w = e & 63;
            corRow[(size_t)q * HW + w] = 0.0f;
        }
        return;
    }

    __shared__ float Bt[104 * BT_STR];         // Bt[s][c], s=0..103 (padded f2 row)
    __shared__ float G[WW * G_STR];            // G[w][s]

    const float* f1row = feats + ((((size_t)b * 8 + t) * CC) * HH + h) * WW;       // +c*HW +w
    const float* f2row = feats + ((((size_t)b * 8 + (t + 1)) * CC) * HH + r) * WW; // +c*HW +w2

    // --- stage Bt: zero pad rows s in [0,20) u [84,104) ---
    for (int idx = tid; idx < 40 * CC; idx += 128) {
        int sz = idx >> 6;                     // 0..39
        int c  = idx & 63;
        int s  = (sz < 20) ? sz : (sz + 64);   // 0..19 , 84..103
        Bt[s * BT_STR + c] = 0.0f;
    }
    // --- stage Bt: data rows s in [20,84) from f2 (coalesced b64, no predication) ---
    for (int idx = tid; idx < CC * 32; idx += 128) {
        int c  = idx >> 5;
        int w2 = (idx & 31) * 2;
        v2f v = *(const v2f*)(f2row + (size_t)c * HW + w2);
        Bt[(w2 + 20) * BT_STR + c] = v.x;
        Bt[(w2 + 21) * BT_STR + c] = v.y;
    }
    __syncthreads();

    const int half = lane >> 4;                // 0 = lanes 0-15, 1 = lanes 16-31
    const int l16  = lane & 15;
    const int wA   = wave * 16 + l16;          // A row (M = w)

    // Preload A k-fragments from global (coalesced, unconditional).
    // Doc layout: lo lanes v0/v1 = K0/K1, hi lanes = K2/K3.
    v2f a[16];
#pragma unroll
    for (int k = 0; k < 16; ++k) {
        const int c0 = 4 * k + 2 * half;
        a[k].x = f1row[(size_t)c0 * HW + wA];
        a[k].y = f1row[(size_t)(c0 + 1) * HW + wA];
    }

    // Only N-tiles j=1..5 are non-trivial (s in [16,96)); branch-free inner loop.
    for (int j = 1; j <= 5; ++j) {
        const int s = 16 * j + l16;            // G column
        const float* brow = &Bt[s * BT_STR];
        v8f acc = {0.f, 0.f, 0.f, 0.f, 0.f, 0.f, 0.f, 0.f};
#pragma unroll
        for (int k = 0; k < 16; ++k) {
            v2f bf = *(const v2f*)(brow + 4 * k + 2 * half);   // ds_load_b64
            acc = __builtin_amdgcn_wmma_f32_16x16x4_f32(
                false, a[k], false, bf, (short)0, acc, false, false);
        }
        // D layout: vgpr i, lanes 0-15 -> M=i, lanes 16-31 -> M=i+8; N = lane%16
#pragma unroll
        for (int i = 0; i < 8; ++i) {
            G[(wave * 16 + i + 8 * half) * G_STR + s] = acc[i];
        }
    }
    __syncthreads();

    // Extraction: cor[n, p*21+q, h, w] = G[w, w+2q]; s outside [16,96) is zero.
    for (int e = tid; e < PATCH * WW; e += 128) {
        int q = e >> 6, w = e & 63;
        int s = w + 2 * q;
        float v = 0.0f;
        if (s >= 16 && s < 96) v = G[w * G_STR + s];
        corRow[(size_t)q * HW + w] = v;
    }
}

// ---------------------------------------------------------------------------
// Kernel 2: per-channel batch statistics (deterministic tree reduction)
// Grid: 441 blocks x 256 threads. meanvar[0..440]=mean, [441..881]=var (biased)
// ---------------------------------------------------------------------------
__global__ __launch_bounds__(256)
void stats_kernel(const float* __restrict__ cor, float* __restrict__ meanvar)
{
    const int pq  = blockIdx.x;
    const int tid = threadIdx.x;
    float s = 0.f, s2 = 0.f;
    for (int n = 0; n < NPAIR; ++n) {
        const float* base = cor + ((size_t)n * PP + pq) * HW;
        for (int i = tid; i < HW; i += 256) {
            float v = base[i];
            s += v;
            s2 += v * v;
        }
    }
    __shared__ float sh[256];
    __shared__ float sh2[256];
    sh[tid] = s; sh2[tid] = s2;
    __syncthreads();
    for (int off = 128; off > 0; off >>= 1) {
        if (tid < off) { sh[tid] += sh[tid + off]; sh2[tid] += sh2[tid + off]; }
        __syncthreads();
    }
    if (tid == 0) {
        const float inv = 1.0f / (float)(NPAIR * HW);
        float m = sh[0] * inv;
        float var = sh2[0] * inv - m * m;
        meanvar[pq]      = m;
        meanvar[PP + pq] = var;
    }
}

// ---------------------------------------------------------------------------
// Kernel 3: fold BN into conv.  Stores W' TRANSPOSED: wprimeT[p][o], p padded to 448.
//           bias[o] = sum_p W[o,p]*(beta - gamma*mean*rsqrt(var+eps))
// Grid: 64 blocks x 256 threads.
// ---------------------------------------------------------------------------
__global__ __launch_bounds__(256)
void prep_kernel(const float* __restrict__ conv_w, const float* __restrict__ gamma,
                 const float* __restrict__ beta, const float* __restrict__ meanvar,
                 float* __restrict__ wprimeT, float* __restrict__ bias)
{
    const int o   = blockIdx.x;
    const int tid = threadIdx.x;
    float acc = 0.f;
    for (int p = tid; p < KP; p += 256) {
        float wv = 0.f;
        if (p < PP) {
            float m   = meanvar[p];
            float var = meanvar[PP + p];
            float rs  = 1.0f / sqrtf(var + 1e-5f);
            float w   = conv_w[o * PP + p];
            wv = w * gamma[p] * rs;
            acc += w * (beta[p] - gamma[p] * m * rs);
        }
        wprimeT[p * CC + o] = wv;
    }
    __shared__ float sh[256];
    sh[tid] = acc;
    __syncthreads();
    for (int off = 128; off > 0; off >>= 1) {
        if (tid < off) sh[tid] += sh[tid + off];
        __syncthreads();
    }
    if (tid == 0) bias[o] = sh[0];
}

// ---------------------------------------------------------------------------
// Kernel 4: zero the appended t=7 frame: out[b, 7, :, :, :] = 0
// ---------------------------------------------------------------------------
__global__ __launch_bounds__(256)
void zero_tail_kernel(float* __restrict__ out)
{
    const int total = BB * CC * HW;            // 1,048,576
    for (int e = blockIdx.x * 256 + threadIdx.x; e < total; e += gridDim.x * 256) {
        int b = e / (CC * HW);
        int rem = e - b * (CC * HW);
        out[(((size_t)b * 8 + 7) * CC) * HW + rem] = 0.0f;
    }
}

// ---------------------------------------------------------------------------
// Kernel 5: 1x1 conv GEMM  out[o,pix] = sum_p W'[o,p]*cor[p,pix] + bias[o]
// Grid: (wtile=4, h=64, n=28), block = 128 (wave m owns output-channel tile m)
// cor pixel-tile staged transposed in LDS (read from HBM exactly once, shared
// by all 4 waves); pad channels zeroed at stage time -> branch-free MAC loop.
// ---------------------------------------------------------------------------
__global__ __launch_bounds__(128)
void conv_wmma_kernel(const float* __restrict__ cor, const float* __restrict__ wprimeT,
                      const float* __restrict__ bias, float* __restrict__ out)
{
    const int wt = blockIdx.x;
    const int h  = blockIdx.y;
    const int n  = blockIdx.z;
    const int b  = n / TT, t = n % TT;
    const int tid  = threadIdx.x;
    const int lane = tid & 31;
    const int wave = tid >> 5;                 // output-channel tile
    const int half = lane >> 4;
    const int l16  = lane & 15;
    const int w0   = wt * 16;

    __shared__ float Bt[16 * CBT_STR];         // Bt[pix][p], p = 0..447

    const float* corBase = cor + (size_t)n * PP * HW + h * WW + w0;   // +p*HW +i

    // zero pad channels p in [441,448)
    if (tid < 112) {
        int i  = tid / 7;
        int pz = PP + (tid % 7);
        Bt[i * CBT_STR + pz] = 0.0f;
    }
    // stage 441 x 16 tile (transposed)
    for (int idx = tid; idx < PP * 16; idx += 128) {
        int pc = idx >> 4, i = idx & 15;
        Bt[i * CBT_STR + pc] = corBase[(size_t)pc * HW + i];
    }
    __syncthreads();

    const int oA = wave * 16 + l16;            // A row (M = out channel)

    v8f acc;
#pragma unroll
    for (int i = 0; i < 8; ++i) acc[i] = bias[wave * 16 + i + 8 * half];

    const float* brow = &Bt[l16 * CBT_STR];
#pragma unroll 4
    for (int k = 0; k < KP / 4; ++k) {         // 112 k-steps over p
        const int p0 = 4 * k + 2 * half;
        v2f af, bf;
        af.x = wprimeT[p0 * CC + oA];          // coalesced b32 (lanes vary in o)
        af.y = wprimeT[(p0 + 1) * CC + oA];
        bf = *(const v2f*)(brow + p0);         // ds_load_b64, conflict-free
        acc = __builtin_amdgcn_wmma_f32_16x16x4_f32(
            false, af, false, bf, (short)0, acc, false, false);
    }

#pragma unroll
    for (int i = 0; i < 8; ++i) {
        const int o = wave * 16 + i + 8 * half;
        out[((((size_t)b * 8 + t) * CC + o) * HW) + h * WW + w0 + l16] = acc[i];
    }
}

// ---------------------------------------------------------------------------
// Host launcher
// ---------------------------------------------------------------------------
extern "C" void kernel_launch(void* const* d_in, const int* in_sizes, int n_in,
                              void* d_out, int out_size, void* d_ws, size_t ws_size,
                              hipStream_t stream)
{
    (void)in_sizes; (void)n_in; (void)out_size; (void)ws_size;

    const float* feats  = (const float*)d_in[0];
    const float* gamma  = (const float*)d_in[1];
    const float* beta   = (const float*)d_in[2];
    const float* conv_w = (const float*)d_in[3];
    float* out = (float*)d_out;

    // Workspace layout (float offsets)
    float* ws       = (float*)d_ws;
    float* cor      = ws;                                  // 28*441*4096 = 50,577,408
    float* meanvar  = cor + (size_t)NPAIR * PP * HW;       // 1024 slot (882 used)
    float* wprimeT  = meanvar + 1024;                      // 448*64 = 28,672
    float* bias     = wprimeT + (size_t)KP * CC;           // 64

    // 1) correlation (WMMA f32 GEMM per (n, dy, h))
    corr_wmma_kernel<<<dim3(HH, PATCH, NPAIR), 128, 0, stream>>>(feats, cor);
    // 2) batch statistics
    stats_kernel<<<dim3(PP), 256, 0, stream>>>(cor, meanvar);
    // 3) fold BN into conv weights (transposed) + bias
    prep_kernel<<<dim3(CC), 256, 0, stream>>>(conv_w, gamma, beta, meanvar, wprimeT, bias);
    // 4) zero the appended time frame
    zero_tail_kernel<<<dim3(512), 256, 0, stream>>>(out);
    // 5) 1x1 conv as WMMA GEMM (K=441), writes frames t=0..6
    conv_wmma_kernel<<<dim3(4, HH, NPAIR), 128, 0, stream>>>(cor, wprimeT, bias, out);
}